// IoULoss_18399639896543
// MI455X (gfx1250) — compile-verified
//
#include <hip/hip_runtime.h>

// ---------------------------------------------------------------------------
// IoU loss, MI455X (gfx1250, wave32).
// Streaming reduction, bandwidth-bound: ~287 MB single pass -> ~12.3 us floor
// at 23.3 TB/s. 128-bit non-temporal loads; v_wmma_f32_16x16x4_f32 used for
// the exact-f32 cross-lane reduction (A = ones, B = lane accumulators).
// ---------------------------------------------------------------------------

#define NUM_CLASSES 13
#define HW_PIX      640000            // 800*800
#define BSZ         8
#define NPIX        (BSZ * HW_PIX)    // 5,120,000
#define NQUAD       (NPIX / 4)        // 1,280,000 (HW_PIX % 4 == 0)
#define NB          2048
#define NT          256
#define NWAVES      (NT / 32)

typedef __attribute__((ext_vector_type(2))) float v2f;
typedef __attribute__((ext_vector_type(4))) float v4f;
typedef __attribute__((ext_vector_type(8))) float v8f;
typedef __attribute__((ext_vector_type(4))) int   v4i;

// Full-wave (32 lane) sum in exact f32.
// WMMA: A = ones(16x4)  => D[m][n] = B[0][n] + B[1][n] + B[2][n] + B[3][n].
// B-VGPR0 holds v[lane]: lanes 0-15 supply K=0 col n, lanes 16-31 supply K=2
// col n-16; B-VGPR1 = 0. So D[m][n] = v[n] + v[n+16] for every row m, and
// element 0 of the D vector in lane L is pairsum[L & 15]. Four xor-shuffle
// adds inside each 16-lane half then produce the 32-lane total in all lanes.
__device__ __forceinline__ float wave_sum32(float v) {
  v2f a; a[0] = 1.0f; a[1] = 1.0f;   // ones A matrix
  v2f b; b[0] = v;    b[1] = 0.0f;   // lane accumulators in K=0 / K=2
  v8f c = {};
  c = __builtin_amdgcn_wmma_f32_16x16x4_f32(false, a, false, b,
                                            (short)0, c, false, false);
  float s = c[0];                     // pairsum[lane & 15], all lanes valid
  s += __shfl_xor(s, 1, 32);
  s += __shfl_xor(s, 2, 32);
  s += __shfl_xor(s, 4, 32);
  s += __shfl_xor(s, 8, 32);
  return s;                           // wave total, replicated to all lanes
}

__global__ __launch_bounds__(NT) void iou_partial_kernel(
    const float* __restrict__ in, const int* __restrict__ tgt,
    float* __restrict__ partial) {
  float acc_i = 0.0f;   // intersection partial
  float acc_t = 0.0f;   // probs.sum() partial
  const int stride = gridDim.x * blockDim.x;
  for (int q = blockIdx.x * blockDim.x + threadIdx.x; q < NQUAD; q += stride) {
    const int p4 = q << 2;                       // first pixel of this quad
    const int b  = p4 / HW_PIX;
    const int hw = p4 - b * HW_PIX;
    const size_t base = (size_t)b * (size_t)(NUM_CLASSES * HW_PIX) + (size_t)hw;

    v4f x[NUM_CLASSES];
#pragma unroll
    for (int c = 0; c < NUM_CLASSES; ++c) {
      // single-pass stream: 16B aligned (HW_PIX % 4 == 0), non-temporal
      x[c] = __builtin_nontemporal_load(
          reinterpret_cast<const v4f*>(in + base + (size_t)c * HW_PIX));
    }
    const v4i t4 = __builtin_nontemporal_load(
        reinterpret_cast<const v4i*>(tgt + p4));

#pragma unroll
    for (int j = 0; j < 4; ++j) {
      const int tj = t4[j];
      float m = x[0][j];
#pragma unroll
      for (int c = 1; c < NUM_CLASSES; ++c) m = fmaxf(m, x[c][j]);
      float s = 0.0f, et = 0.0f;
#pragma unroll
      for (int c = 0; c < NUM_CLASSES; ++c) {
        const float e = __expf(x[c][j] - m);
        s += e;
        et = (c == tj) ? e : et;
      }
      const float inv = __builtin_amdgcn_rcpf(s);
      acc_i += et * inv;
      acc_t += s * inv;   // numerically-computed softmax row sum (~1)
    }
  }

  // EXEC is all-ones here (uniform path after the loop) -> WMMA legal.
  const float wi = wave_sum32(acc_i);
  const float wt = wave_sum32(acc_t);

  __shared__ float sm[2 * NWAVES];
  const int lane = threadIdx.x & 31;
  const int wid  = threadIdx.x >> 5;
  if (lane == 0) { sm[wid] = wi; sm[NWAVES + wid] = wt; }
  __syncthreads();
  if (threadIdx.x == 0) {
    float bi = 0.0f, bt = 0.0f;
#pragma unroll
    for (int w = 0; w < NWAVES; ++w) { bi += sm[w]; bt += sm[NWAVES + w]; }
    partial[2 * blockIdx.x + 0] = bi;     // plain store: deterministic
    partial[2 * blockIdx.x + 1] = bt;
  }
}

__global__ __launch_bounds__(NT) void iou_final_kernel(
    const float* __restrict__ partial, const void* __restrict__ smooth_p,
    float* __restrict__ out) {
  float ai = 0.0f, at = 0.0f;
  for (int i = threadIdx.x; i < NB; i += NT) {
    ai += partial[2 * i + 0];
    at += partial[2 * i + 1];
  }
  const float wi = wave_sum32(ai);
  const float wt = wave_sum32(at);

  __shared__ float sm[2 * NWAVES];
  const int lane = threadIdx.x & 31;
  const int wid  = threadIdx.x >> 5;
  if (lane == 0) { sm[wid] = wi; sm[NWAVES + wid] = wt; }
  __syncthreads();
  if (threadIdx.x == 0) {
    float I = 0.0f, T = 0.0f;
#pragma unroll
    for (int w = 0; w < NWAVES; ++w) { I += sm[w]; T += sm[NWAVES + w]; }
    // smooth may be stored as integer bits or as float bits; small-magnitude
    // bit patterns are integers, anything else is an f32 payload.
    const unsigned u = *reinterpret_cast<const unsigned*>(smooth_p);
    const float smooth = (u < 0x01000000u) ? (float)(int)u
                                           : __uint_as_float(u);
    const float one_hot_sum = (float)NPIX;       // targets.size
    const float total = T + one_hot_sum;
    const float uni   = total - I;
    out[0] = 1.0f - (I + smooth) / (uni + smooth);
  }
}

extern "C" void kernel_launch(void* const* d_in, const int* in_sizes, int n_in,
                              void* d_out, int out_size, void* d_ws,
                              size_t ws_size, hipStream_t stream) {
  const float* in   = (const float*)d_in[0];
  const int*   tgt  = (const int*)d_in[1];
  const void*  smo  = d_in[2];
  float* partial = (float*)d_ws;                 // 2 * NB floats = 16 KB

  iou_partial_kernel<<<NB, NT, 0, stream>>>(in, tgt, partial);
  iou_final_kernel<<<1, NT, 0, stream>>>(partial, smo, (float*)d_out);
}